// CapsuleLayer_19224273617143
// MI455X (gfx1250) — compile-verified
//
#include <hip/hip_runtime.h>

typedef __attribute__((ext_vector_type(2))) float v2f;
typedef __attribute__((ext_vector_type(8))) float v8f;

#define B_    128
#define I_    1152
#define P_    8
#define J_    10
#define D_    16
#define JD_   160          // J_*D_
#define CHUNK 144          // I_ / 8 chunks per batch for the s-reduction
#define NCHK  8

// ---------------------------------------------------------------------------
// Kernel 1: inputs_hat[b,i,jd] = sum_p W[i,jd,p] * x[b,i,p]  via f32 WMMA.
// One wave handles one (i, 16-wide b-tile): D[160x16] = A[160x8] x B[8x16]
// as 10 M-tiles x 2 K-steps of v_wmma_f32_16x16x4_f32.
// ---------------------------------------------------------------------------
__global__ __launch_bounds__(256) void caps_hat_wmma(
    const float* __restrict__ x,    // [B, I, P]
    const float* __restrict__ W,    // [I, JD, P]  (== W[0,i,j,d,p])
    float* __restrict__ hat)        // [B, I, JD]
{
    const int lane   = threadIdx.x & 31;
    const int wave   = threadIdx.x >> 5;
    const int i      = blockIdx.x * 8 + wave;   // [0, 1152)
    const int b_base = blockIdx.y * 16;         // [0, 128) step 16
    const int half   = lane >> 4;               // 0: lanes 0-15, 1: lanes 16-31
    const int lm     = lane & 15;
    const int kb     = half * 2;                // K pair {0,1} or {2,3}

    // B operand (x): lane holds column N = b_base+lm; reused for all 10 M-tiles.
    const float* xp = x + (size_t)(b_base + lm) * (I_ * P_) + (size_t)i * P_;
    v2f b0 = *(const v2f*)(xp + kb);        // k = kb, kb+1
    v2f b1 = *(const v2f*)(xp + kb + 4);    // k = kb+4, kb+5

    const float* wp = W + (size_t)i * (JD_ * P_);

    #pragma unroll
    for (int mt = 0; mt < 10; ++mt) {
        const int row = mt * 16 + lm;       // jd row of A (same row in both halves)
        v2f a0 = *(const v2f*)(wp + row * P_ + kb);
        v2f a1 = *(const v2f*)(wp + row * P_ + kb + 4);
        v8f c = {};
        c = __builtin_amdgcn_wmma_f32_16x16x4_f32(false, a0, false, b0,
                                                  (short)0, c, false, false);
        c = __builtin_amdgcn_wmma_f32_16x16x4_f32(false, a1, false, b1,
                                                  (short)0, c, false, false);
        // Lane owns column b = b_base+lm; rows jd = mt*16 + half*8 + r, r=0..7
        float* dst = hat + ((size_t)(b_base + lm) * I_ + i) * JD_ + mt * 16 + half * 8;
        *(float4*)(dst + 0) = make_float4(c[0], c[1], c[2], c[3]);
        *(float4*)(dst + 4) = make_float4(c[4], c[5], c[6], c[7]);
    }
}

// ---------------------------------------------------------------------------
// R1: softmax of logits over J, then partial s[b,j,d] over a chunk of 144 i's.
// Block = 160 threads (5 waves); grid = (B, NCHK). Deterministic (no atomics).
// ---------------------------------------------------------------------------
__global__ __launch_bounds__(160) void caps_route_partial(
    const float* __restrict__ blog,   // [B, I, J]
    const float* __restrict__ hat,    // [B, I, JD]
    float* __restrict__ partial,      // [B, NCHK, JD]
    int uniform)
{
    __shared__ float cLDS[CHUNK * J_];
    const int b  = blockIdx.x;
    const int ch = blockIdx.y;
    const int i0 = ch * CHUNK;
    const int t  = threadIdx.x;

    if (t < CHUNK) {
        if (uniform) {
            #pragma unroll
            for (int j = 0; j < J_; ++j) cLDS[t * J_ + j] = 1.0f / (float)J_;
        } else {
            const float* bp = blog + ((size_t)b * I_ + i0 + t) * J_;
            float v[J_];
            float m = -1e30f;
            #pragma unroll
            for (int j = 0; j < J_; ++j) { v[j] = bp[j]; m = fmaxf(m, v[j]); }
            float sum = 0.0f;
            #pragma unroll
            for (int j = 0; j < J_; ++j) { v[j] = expf(v[j] - m); sum += v[j]; }
            const float inv = 1.0f / sum;
            #pragma unroll
            for (int j = 0; j < J_; ++j) cLDS[t * J_ + j] = v[j] * inv;
        }
    }
    __syncthreads();

    const int j = t >> 4;   // t in [0,160): jd index; j = t/16
    float acc = 0.0f;
    const float* hp = hat + ((size_t)b * I_ + i0) * JD_ + t;
    #pragma unroll 4
    for (int i = 0; i < CHUNK; ++i)
        acc += cLDS[i * J_ + j] * hp[(size_t)i * JD_];   // coalesced 640B rows
    partial[((size_t)b * NCHK + ch) * JD_ + t] = acc;
}

// ---------------------------------------------------------------------------
// R2: reduce chunk partials, squash, write outputs [B, J, D] (= d_out).
// ---------------------------------------------------------------------------
__global__ __launch_bounds__(160) void caps_squash(
    const float* __restrict__ partial,  // [B, NCHK, JD]
    float* __restrict__ out)            // [B, JD]
{
    __shared__ float sl[JD_];
    const int b = blockIdx.x;
    const int t = threadIdx.x;

    float s = 0.0f;
    #pragma unroll
    for (int ch = 0; ch < NCHK; ++ch)
        s += partial[((size_t)b * NCHK + ch) * JD_ + t];
    sl[t] = s;
    __syncthreads();

    const int j = t >> 4;
    float s2 = 0.0f;
    #pragma unroll
    for (int d = 0; d < D_; ++d) { float v = sl[j * D_ + d]; s2 += v * v; }
    const float scale = (s2 / (1.0f + s2)) * rsqrtf(s2 + 1e-7f);
    out[(size_t)b * JD_ + t] = scale * s;
}

// ---------------------------------------------------------------------------
// R3: agreement[b,i,j] = <hat[b,i,j,:], out[b,j,:]>; logits = (add? logits:0)+agr
// One thread per (b,i,j); float4 dot products (hat row is 64B aligned).
// ---------------------------------------------------------------------------
__global__ __launch_bounds__(256) void caps_agree(
    const float* __restrict__ hat,   // [B, I, JD]
    const float* __restrict__ out,   // [B, JD]
    float* __restrict__ blog,        // [B, I, J]
    int add)
{
    const int tid = blockIdx.x * blockDim.x + threadIdx.x;
    const int total = B_ * I_ * J_;
    if (tid >= total) return;
    const int j  = tid % J_;
    const int bi = tid / J_;          // b*I_ + i
    const int b  = bi / I_;

    const float4* hp = (const float4*)(hat + (size_t)bi * JD_ + j * D_);
    const float4* op = (const float4*)(out + (size_t)b * JD_ + j * D_);
    float acc = 0.0f;
    #pragma unroll
    for (int q = 0; q < 4; ++q) {
        float4 h = hp[q], o = op[q];
        acc += h.x * o.x + h.y * o.y + h.z * o.z + h.w * o.w;
    }
    const size_t bidx = (size_t)bi * J_ + j;
    blog[bidx] = add ? (blog[bidx] + acc) : acc;
}

// ---------------------------------------------------------------------------
extern "C" void kernel_launch(void* const* d_in, const int* in_sizes, int n_in,
                              void* d_out, int out_size, void* d_ws, size_t ws_size,
                              hipStream_t stream) {
    const float* x = (const float*)d_in[0];   // [128,1152,8]
    const float* W = (const float*)d_in[1];   // [1,1152,10,16,8]
    float* out = (float*)d_out;               // [128,10,16]

    float* ws      = (float*)d_ws;
    float* hat     = ws;                                   // 23,592,960 f
    float* blog    = hat + (size_t)B_ * I_ * JD_;          //  1,474,560 f
    float* partial = blog + (size_t)B_ * I_ * J_;          //    163,840 f

    // 1) inputs_hat via f32 WMMA
    caps_hat_wmma<<<dim3(I_ / 8, B_ / 16), 256, 0, stream>>>(x, W, hat);

    const int agr_blocks = (B_ * I_ * J_ + 255) / 256;     // 5760

    // routing iteration 0 (uniform coupling; logits written fresh, never read)
    caps_route_partial<<<dim3(B_, NCHK), 160, 0, stream>>>(blog, hat, partial, 1);
    caps_squash<<<B_, 160, 0, stream>>>(partial, out);
    caps_agree<<<agr_blocks, 256, 0, stream>>>(hat, out, blog, 0);

    // routing iteration 1
    caps_route_partial<<<dim3(B_, NCHK), 160, 0, stream>>>(blog, hat, partial, 0);
    caps_squash<<<B_, 160, 0, stream>>>(partial, out);
    caps_agree<<<agr_blocks, 256, 0, stream>>>(hat, out, blog, 1);

    // routing iteration 2 (final squash -> d_out)
    caps_route_partial<<<dim3(B_, NCHK), 160, 0, stream>>>(blog, hat, partial, 0);
    caps_squash<<<B_, 160, 0, stream>>>(partial, out);
}